// GRUModel_6914897346748
// MI455X (gfx1250) — compile-verified
//
#include <hip/hip_runtime.h>
#include <stdint.h>

// GRU: B=2048, T=1024, D=6, H=48, G=3H=144.
// One wave (32 threads) per 16-batch-row tile; 128 blocks total.
// Recurrence kept fully in fp32 via V_WMMA_F32_16X16X4_F32:
//   M = batch tile (16), N = gate columns (9 tiles of 16), K = H (12 chunks of 4).
// x_t staged to LDS via async global->LDS loads, double buffered.

#define TT 1024
#define DM 6
#define HH 48
#define GG 144

typedef __attribute__((ext_vector_type(2))) float v2f;
typedef __attribute__((ext_vector_type(8))) float v8f;

__device__ __forceinline__ v8f splat8(float v) {
  v8f r;
#pragma unroll
  for (int i = 0; i < 8; ++i) r[i] = v;
  return r;
}

__device__ __forceinline__ v8f wmma4(v2f a, v2f b, v8f c) {
  // D = A(16x4 f32) * B(4x16 f32) + C(16x16 f32)
  return __builtin_amdgcn_wmma_f32_16x16x4_f32(false, a, false, b, (short)0, c,
                                               false, false);
}

__device__ __forceinline__ float sigmoid_(float x) {
  return __builtin_amdgcn_rcpf(1.0f + __expf(-x));
}

__device__ __forceinline__ float tanh_(float x) {
#if __has_builtin(__builtin_amdgcn_tanhf)
  return __builtin_amdgcn_tanhf(x);
#else
  float e = __expf(2.0f * x);
  return 1.0f - 2.0f * __builtin_amdgcn_rcpf(e + 1.0f);
#endif
}

__device__ __forceinline__ uint32_t lds_off(const void* p) {
  return (uint32_t)(uintptr_t)(__attribute__((address_space(3))) const void*)p;
}

__device__ __forceinline__ void async_ld_b32(uint32_t lds_addr, const float* g) {
  asm volatile("global_load_async_to_lds_b32 %0, %1, off"
               :: "v"(lds_addr), "v"((uint64_t)(uintptr_t)g)
               : "memory");
}

__global__ __launch_bounds__(32) void gru_fused_kernel(
    const float* __restrict__ x, const float* __restrict__ W_ih,
    const float* __restrict__ W_hh, const float* __restrict__ b_ih,
    const float* __restrict__ b_hh, const float* __restrict__ fc_w,
    const float* __restrict__ fc_b, float* __restrict__ out) {
  // Pair-interleaved transposed weights: s_whh[kp][g][j] = W_hh[g][2*kp+j]
  // so a B operand (VGPR0=row 4k+2h, VGPR1=row 4k+2h+1) is one ds_load_b64.
  __shared__ __align__(16) float s_whh[24][GG][2];   // 27.6 KB
  __shared__ __align__(16) float s_wih[4][GG][2];    // K padded 6->8, rows 6,7 = 0
  __shared__ __align__(16) float s_fcw[24][16][2];   // N padded 6->16 with zeros
  __shared__ __align__(16) float s_h[16][HH];        // hidden state (row-major)
  __shared__ __align__(16) float s_x[2][16][8];      // x_t double buffer, padded to 8

  const int tid = threadIdx.x;
  const int col = tid & 15;   // N index within a 16-wide tile / A-row (M=l%16)
  const int half = tid >> 4;  // lane half selects K parity (A/B) and M+8 (C)
  const int bBase = blockIdx.x * 16;

  // ---- stage weights into LDS (single wave; DS ops are in-order) ----
  for (int idx = tid; idx < 24 * GG * 2; idx += 32) {
    int kp = idx / (GG * 2), r = idx % (GG * 2), g = r >> 1, j = r & 1;
    s_whh[kp][g][j] = W_hh[g * HH + 2 * kp + j];
  }
  for (int idx = tid; idx < 4 * GG * 2; idx += 32) {
    int kp = idx / (GG * 2), r = idx % (GG * 2), g = r >> 1, j = r & 1;
    int k = 2 * kp + j;
    s_wih[kp][g][j] = (k < DM) ? W_ih[g * DM + k] : 0.0f;
  }
  for (int idx = tid; idx < 24 * 16 * 2; idx += 32) {
    int kp = idx / 32, r = idx % 32, d = r >> 1, j = r & 1;
    s_fcw[kp][d][j] = (d < DM) ? fc_w[d * HH + 2 * kp + j] : 0.0f;
  }
  for (int idx = tid; idx < 16 * HH; idx += 32) (&s_h[0][0])[idx] = 0.0f;
  for (int idx = tid; idx < 2 * 16 * 8; idx += 32) (&s_x[0][0][0])[idx] = 0.0f;

  // ---- per-lane biases (gate g = 16*tile + col) ----
  float brz[6], bni[3], bnh[3];
#pragma unroll
  for (int n = 0; n < 6; ++n)
    brz[n] = b_ih[16 * n + col] + b_hh[16 * n + col];
#pragma unroll
  for (int j = 0; j < 3; ++j) {
    bni[j] = b_ih[96 + 16 * j + col];
    bnh[j] = b_hh[96 + 16 * j + col];
  }
  const float fcb = (col < DM) ? fc_b[col] : 0.0f;

  // hidden state in registers, C layout: hC[j][i] = h[b = i+8*half][16*j+col]
  v8f hC[3];
#pragma unroll
  for (int j = 0; j < 3; ++j) hC[j] = splat8(0.0f);

  // ---- prefetch x for t=0 into buffer 0 (3 async b32 per lane = 96 dwords) ----
#pragma unroll
  for (int q = 0; q < 3; ++q) {
    int e = tid + 32 * q, row = e / DM, d = e % DM;
    async_ld_b32(lds_off(&s_x[0][row][d]),
                 x + ((size_t)(bBase + row) * TT + 0) * DM + d);
  }

  for (int t = 0; t < TT; ++t) {
    const int buf = t & 1;
    if (t + 1 < TT) {
#pragma unroll
      for (int q = 0; q < 3; ++q) {
        int e = tid + 32 * q, row = e / DM, d = e % DM;
        async_ld_b32(lds_off(&s_x[buf ^ 1][row][d]),
                     x + ((size_t)(bBase + row) * TT + (t + 1)) * DM + d);
      }
      asm volatile("s_wait_asynccnt 3" ::: "memory");  // t's 3 loads done
    } else {
      asm volatile("s_wait_asynccnt 0" ::: "memory");
    }

    // ---- A operands (lane = row M, VGPR0/1 = K 2*half / 2*half+1) ----
    v2f Ax[2], Ah[12];
#pragma unroll
    for (int k = 0; k < 2; ++k)
      Ax[k] = *(const v2f*)&s_x[buf][col][4 * k + 2 * half];
#pragma unroll
    for (int k = 0; k < 12; ++k)
      Ah[k] = *(const v2f*)&s_h[col][4 * k + 2 * half];

    // ---- gate GEMMs: r,z fused (bias + x-part + h-part); n split ----
    v8f accRZ[6], accNI[3], accNH[3];
#pragma unroll
    for (int n = 0; n < 6; ++n) {
      v8f a = splat8(brz[n]);
#pragma unroll
      for (int k = 0; k < 2; ++k)
        a = wmma4(Ax[k], *(const v2f*)&s_wih[2 * k + half][16 * n + col][0], a);
#pragma unroll
      for (int k = 0; k < 12; ++k)
        a = wmma4(Ah[k], *(const v2f*)&s_whh[2 * k + half][16 * n + col][0], a);
      accRZ[n] = a;
    }
#pragma unroll
    for (int j = 0; j < 3; ++j) {
      const int g = 96 + 16 * j + col;
      v8f ai = splat8(bni[j]);
#pragma unroll
      for (int k = 0; k < 2; ++k)
        ai = wmma4(Ax[k], *(const v2f*)&s_wih[2 * k + half][g][0], ai);
      accNI[j] = ai;
      v8f ah = splat8(bnh[j]);
#pragma unroll
      for (int k = 0; k < 12; ++k)
        ah = wmma4(Ah[k], *(const v2f*)&s_whh[2 * k + half][g][0], ah);
      accNH[j] = ah;
    }

    // ---- elementwise GRU update; write h back to LDS for next-step A ----
#pragma unroll
    for (int j = 0; j < 3; ++j) {
#pragma unroll
      for (int i = 0; i < 8; ++i) {
        float r = sigmoid_(accRZ[j][i]);
        float z = sigmoid_(accRZ[3 + j][i]);
        float n = tanh_(accNI[j][i] + r * accNH[j][i]);
        float h = (1.0f - z) * n + z * hC[j][i];
        hC[j][i] = h;
        s_h[i + 8 * half][16 * j + col] = h;  // C layout -> row-major LDS
      }
    }

    // ---- fused output FC: y_t = h_new @ fc_w^T (+ fc_b), via LDS reload ----
    v8f yacc = splat8(fcb);
#pragma unroll
    for (int k = 0; k < 12; ++k) {
      v2f a = *(const v2f*)&s_h[col][4 * k + 2 * half];
      v2f b = *(const v2f*)&s_fcw[2 * k + half][col][0];
      yacc = wmma4(a, b, yacc);
    }
    if (col < DM) {
#pragma unroll
      for (int i = 0; i < 8; ++i)
        out[((size_t)(bBase + i + 8 * half) * TT + t) * DM + col] = yacc[i];
    }
  }
}

extern "C" void kernel_launch(void* const* d_in, const int* in_sizes, int n_in,
                              void* d_out, int out_size, void* d_ws,
                              size_t ws_size, hipStream_t stream) {
  const float* x = (const float*)d_in[0];
  const float* W_ih = (const float*)d_in[1];
  const float* W_hh = (const float*)d_in[2];
  const float* b_ih = (const float*)d_in[3];
  const float* b_hh = (const float*)d_in[4];
  const float* fc_w = (const float*)d_in[5];
  const float* fc_b = (const float*)d_in[6];
  float* out = (float*)d_out;

  const int B = in_sizes[0] / (TT * DM);  // 2048
  dim3 grid(B / 16), block(32);
  gru_fused_kernel<<<grid, block, 0, stream>>>(x, W_ih, W_hh, b_ih, b_hh, fc_w,
                                               fc_b, out);
}